// multilayer_message_passing_88244398064001
// MI455X (gfx1250) — compile-verified
//
#include <hip/hip_runtime.h>
#include <hip/hip_bf16.h>

#define N_GENE 100000
#define N_DIS  50000
#define DIM    128
#define NEDGE  800000

typedef float v2f __attribute__((ext_vector_type(2)));
typedef float v8f __attribute__((ext_vector_type(8)));

// ---------------------------------------------------------------------------
// Scatter: sum[dst] += x_src[src], cnt[dst] += 1   (mean aggregation numerator)
// One thread per (edge, float4 chunk): 32 threads per edge.
// ---------------------------------------------------------------------------
__global__ __launch_bounds__(256) void sage_scatter(
    const float* __restrict__ x_src, const int* __restrict__ src,
    const int* __restrict__ dst, float* __restrict__ sum,
    float* __restrict__ cnt, int nedge)
{
    unsigned long long gid = (unsigned long long)blockIdx.x * blockDim.x + threadIdx.x;
    int e = (int)(gid >> 5);
    int q = (int)(gid & 31);
    if (e >= nedge) return;
    int s = src[e];
    int d = dst[e];
    const float4 v = ((const float4*)x_src)[(size_t)s * (DIM / 4) + q];
    float* p = sum + (size_t)d * DIM + q * 4;
    __hip_atomic_fetch_add(p + 0, v.x, __ATOMIC_RELAXED, __HIP_MEMORY_SCOPE_AGENT);
    __hip_atomic_fetch_add(p + 1, v.y, __ATOMIC_RELAXED, __HIP_MEMORY_SCOPE_AGENT);
    __hip_atomic_fetch_add(p + 2, v.z, __ATOMIC_RELAXED, __HIP_MEMORY_SCOPE_AGENT);
    __hip_atomic_fetch_add(p + 3, v.w, __ATOMIC_RELAXED, __HIP_MEMORY_SCOPE_AGENT);
    if (q == 0)
        __hip_atomic_fetch_add(cnt + d, 1.0f, __ATOMIC_RELAXED, __HIP_MEMORY_SCOPE_AGENT);
}

// ---------------------------------------------------------------------------
// Fused SAGE layer: h = (sum/max(cnt,1)) @ Wl^T + x_old @ Wr^T + b
//                   [relu] ; h /= max(||h||2, 1e-12) ; out = h + x_old
// 256 threads = 8 waves; wave w handles rows [blk*128 + w*16, +16), all 128 cols.
// Uses V_WMMA_F32_16X16X4_F32 (fp32 exact). Weights staged via LDS.
// ---------------------------------------------------------------------------
__global__ __launch_bounds__(256) void sage_gemm(
    const float* __restrict__ sum, const float* __restrict__ cnt,
    const float* xold,
    const float* __restrict__ Wl, const float* __restrict__ Wr,
    const float* __restrict__ bias,
    float* out, int n, int do_relu)
{
    __shared__ float sW[DIM * DIM];          // 64 KB weight stage

    const int tid  = threadIdx.x;
    const int wave = tid >> 5;
    const int lane = tid & 31;
    const int half = lane >> 4;              // 0: lanes 0-15, 1: lanes 16-31
    const int jb   = lane & 15;              // column-in-tile / row-in-16 selector
    const int koff = half << 1;              // K sub-offset for A/B fragments

    const int r0   = blockIdx.x * 128 + wave * 16;
    int arow = r0 + jb;                      // A-fragment row for this lane
    if (arow > n - 1) arow = n - 1;          // clamp (stores are predicated later)

    // stage Wl into LDS
    for (int i = tid; i < DIM * DIM / 4; i += 256)
        ((float4*)sW)[i] = ((const float4*)Wl)[i];
    __syncthreads();

    const float rcp = 1.0f / fmaxf(cnt[arow], 1.0f);

    v8f acc[8];
    const v8f zero = {0.f, 0.f, 0.f, 0.f, 0.f, 0.f, 0.f, 0.f};
#pragma unroll
    for (int t = 0; t < 8; ++t) acc[t] = zero;

    // ---- pass 1: mean @ Wl^T  (mean division folded into A fragment) ----
    {
        const v2f* As = (const v2f*)(sum + (size_t)arow * DIM);
#pragma unroll 4
        for (int k = 0; k < DIM; k += 4) {
            v2f a = As[(k + koff) >> 1];
            a = a * rcp;
#pragma unroll
            for (int t = 0; t < 8; ++t) {
                const v2f bf = *(const v2f*)(&sW[(t * 16 + jb) * DIM + k + koff]);
                acc[t] = __builtin_amdgcn_wmma_f32_16x16x4_f32(
                    false, a, false, bf, (short)0, acc[t], false, false);
            }
        }
    }

    // restage Wr
    __syncthreads();
    for (int i = tid; i < DIM * DIM / 4; i += 256)
        ((float4*)sW)[i] = ((const float4*)Wr)[i];
    __syncthreads();

    // ---- pass 2: x_old @ Wr^T (same accumulators -> fused K=256 chain) ----
    {
        const v2f* Xs = (const v2f*)(xold + (size_t)arow * DIM);
#pragma unroll 4
        for (int k = 0; k < DIM; k += 4) {
            v2f a = Xs[(k + koff) >> 1];
#pragma unroll
            for (int t = 0; t < 8; ++t) {
                const v2f bf = *(const v2f*)(&sW[(t * 16 + jb) * DIM + k + koff]);
                acc[t] = __builtin_amdgcn_wmma_f32_16x16x4_f32(
                    false, a, false, bf, (short)0, acc[t], false, false);
            }
        }
    }

    // ---- epilogue: bias, relu, L2-normalize rows, skip-sum, store ----
    // D layout: lane holds col n = t*16 + jb, rows m = v + 8*half (v = VGPR idx)
    float bv[8];
#pragma unroll
    for (int t = 0; t < 8; ++t) bv[t] = bias[t * 16 + jb];

#pragma unroll
    for (int t = 0; t < 8; ++t)
#pragma unroll
        for (int v = 0; v < 8; ++v) {
            float h = acc[t][v] + bv[t];
            if (do_relu) h = fmaxf(h, 0.0f);
            acc[t][v] = h;
        }

    float s2[8];
#pragma unroll
    for (int v = 0; v < 8; ++v) {
        float s = 0.0f;
#pragma unroll
        for (int t = 0; t < 8; ++t) s += acc[t][v] * acc[t][v];
        s2[v] = s;
    }
    // reduce across the 16 lanes sharing a row (xor masks 1,2,4,8 stay in-half)
#pragma unroll
    for (int v = 0; v < 8; ++v) {
        s2[v] += __shfl_xor(s2[v], 1, 32);
        s2[v] += __shfl_xor(s2[v], 2, 32);
        s2[v] += __shfl_xor(s2[v], 4, 32);
        s2[v] += __shfl_xor(s2[v], 8, 32);
    }

    const int rbase = r0 + (half << 3);
#pragma unroll
    for (int v = 0; v < 8; ++v) {
        const int row = rbase + v;
        const float inv = 1.0f / fmaxf(sqrtf(s2[v]), 1e-12f);
        if (row < n) {
#pragma unroll
            for (int t = 0; t < 8; ++t) {
                const size_t idx = (size_t)row * DIM + t * 16 + jb;
                out[idx] = acc[t][v] * inv + xold[idx];
            }
        }
    }
}

// ---------------------------------------------------------------------------
extern "C" void kernel_launch(void* const* d_in, const int* in_sizes, int n_in,
                              void* d_out, int out_size, void* d_ws, size_t ws_size,
                              hipStream_t stream) {
    const float* x_gene = (const float*)d_in[0];
    const float* x_dis  = (const float*)d_in[1];
    const float* Wl     = (const float*)d_in[2];   // (2,2,128,128)
    const float* Wr     = (const float*)d_in[3];   // (2,2,128,128)
    const float* bb     = (const float*)d_in[4];   // (2,2,128)
    const int* src_g2d  = (const int*)d_in[5];
    const int* dst_g2d  = (const int*)d_in[6];
    const int* src_d2g  = (const int*)d_in[7];
    const int* dst_d2g  = (const int*)d_in[8];
    const int nedge = in_sizes[5];

    float* xg = (float*)d_out;                         // N_GENE x 128
    float* xd = xg + (size_t)N_GENE * DIM;             // N_DIS  x 128

    float* ws    = (float*)d_ws;
    float* sum_g = ws;
    float* sum_d = sum_g + (size_t)N_GENE * DIM;
    float* cnt_g = sum_d + (size_t)N_DIS * DIM;
    float* cnt_d = cnt_g + N_GENE;
    const size_t zbytes =
        ((size_t)N_GENE * DIM + (size_t)N_DIS * DIM + N_GENE + N_DIS) * sizeof(float);

    const dim3 blk(256);
    const int scat_blocks = (int)(((long long)nedge * 32 + 255) / 256);
    const int gblk_d = (N_DIS + 127) / 128;
    const int gblk_g = (N_GENE + 127) / 128;

#define WOFF(i, et) ((size_t)((i) * 2 + (et)) * DIM * DIM)
#define BOFF(i, et) ((size_t)((i) * 2 + (et)) * DIM)

    // ---------------- layer 0 (inputs -> d_out regions) ----------------
    hipMemsetAsync(d_ws, 0, zbytes, stream);
    sage_scatter<<<scat_blocks, blk, 0, stream>>>(x_gene, src_g2d, dst_g2d, sum_d, cnt_d, nedge);
    sage_scatter<<<scat_blocks, blk, 0, stream>>>(x_dis,  src_d2g, dst_d2g, sum_g, cnt_g, nedge);
    sage_gemm<<<gblk_d, blk, 0, stream>>>(sum_d, cnt_d, x_dis,
        Wl + WOFF(0, 0), Wr + WOFF(0, 0), bb + BOFF(0, 0), xd, N_DIS, 1);
    sage_gemm<<<gblk_g, blk, 0, stream>>>(sum_g, cnt_g, x_gene,
        Wl + WOFF(0, 1), Wr + WOFF(0, 1), bb + BOFF(0, 1), xg, N_GENE, 1);

    // ---------------- layer 1 (d_out in-place) ----------------
    hipMemsetAsync(d_ws, 0, zbytes, stream);
    sage_scatter<<<scat_blocks, blk, 0, stream>>>(xg, src_g2d, dst_g2d, sum_d, cnt_d, nedge);
    sage_scatter<<<scat_blocks, blk, 0, stream>>>(xd, src_d2g, dst_d2g, sum_g, cnt_g, nedge);
    sage_gemm<<<gblk_d, blk, 0, stream>>>(sum_d, cnt_d, xd,
        Wl + WOFF(1, 0), Wr + WOFF(1, 0), bb + BOFF(1, 0), xd, N_DIS, 0);
    sage_gemm<<<gblk_g, blk, 0, stream>>>(sum_g, cnt_g, xg,
        Wl + WOFF(1, 1), Wr + WOFF(1, 1), bb + BOFF(1, 1), xg, N_GENE, 0);

#undef WOFF
#undef BOFF
}